// GCN2_83270825935314
// MI455X (gfx1250) — compile-verified
//
#include <hip/hip_runtime.h>

typedef float v2f __attribute__((ext_vector_type(2)));
typedef float v8f __attribute__((ext_vector_type(8)));

#define TK 64  // K-chunk staged in LDS per iteration

// ---------------------------------------------------------------------------
// WMMA f32 GEMM:  H[M,Nout] = X[M,K] @ Wp[n][k]^T  (Wp zero-padded 256 x Kpad)
// One block = 16 output rows. 8 waves; wave w owns col-tiles w and w+8.
// Inner loop is branch-free: A fragment from LDS (ds_load_b64), B fragment
// from padded weights (global_load_b64), unconditional v_wmma. EXEC all-1s.
// ---------------------------------------------------------------------------
__global__ __launch_bounds__(256) void k_gemm_wmma(
    const float* __restrict__ X, const float* __restrict__ Wp,
    float* __restrict__ H, int M, int K, int Kpad, int Nout)
{
    __shared__ float As[16][TK + 4];   // +4 pad: conflict-free fragment reads

    const int row0 = blockIdx.x << 4;
    const int t    = threadIdx.x;
    const int wave = t >> 5;
    const int lane = t & 31;
    const int half = lane >> 4;        // 0: K pair {0,1}; 1: K pair {2,3}
    const int m    = lane & 15;

    const int n0 = (wave << 4) + m;          // col tile wave
    const int n1 = ((wave + 8) << 4) + m;    // col tile wave+8 (may be padding)
    const float* __restrict__ Wr0 = Wp + (size_t)n0 * Kpad;
    const float* __restrict__ Wr1 = Wp + (size_t)n1 * Kpad;

    v8f acc0 = {}; v8f acc1 = {};

    for (int k0 = 0; k0 < Kpad; k0 += TK) {
        __syncthreads();
        #pragma unroll
        for (int i = 0; i < (16 * TK) / 256; ++i) {   // 4 elems/thread
            int e  = t + (i << 8);
            int r  = e >> 6;          // e / TK
            int kk = e & (TK - 1);    // e % TK
            int gr = row0 + r, gk = k0 + kk;
            As[r][kk] = (gr < M && gk < K) ? X[(size_t)gr * K + gk] : 0.0f;
        }
        __syncthreads();

        #pragma unroll
        for (int kk = 0; kk < TK; kk += 4) {
            const int kb = kk + (half << 1);          // even -> 8B aligned
            v2f a  = *(const v2f*)&As[m][kb];
            v2f b0 = *(const v2f*)&Wr0[k0 + kb];
            v2f b1 = *(const v2f*)&Wr1[k0 + kb];
            acc0 = __builtin_amdgcn_wmma_f32_16x16x4_f32(
                false, a, false, b0, (short)0, acc0, false, false);
            acc1 = __builtin_amdgcn_wmma_f32_16x16x4_f32(
                false, a, false, b1, (short)0, acc1, false, false);
        }
    }

    // C/D layout: VGPR r, lane L -> D[r + 8*(L/16)][L%16]
    #pragma unroll
    for (int r = 0; r < 8; ++r) {
        int grow = row0 + r + (half << 3);
        if (grow < M) {
            if (n0 < Nout) H[(size_t)grow * Nout + n0] = acc0[r];
            if (n1 < Nout) H[(size_t)grow * Nout + n1] = acc1[r];
        }
    }
}

// ---------------------------------------------------------------------------
// Zero-pad W[Nout,K] -> Wp[256,Kpad]
// ---------------------------------------------------------------------------
__global__ void k_padW(const float* __restrict__ W, float* __restrict__ Wp,
                       int Nout, int K, int Kpad)
{
    int idx = blockIdx.x * blockDim.x + threadIdx.x;
    int tot = 256 * Kpad;
    if (idx >= tot) return;
    int r = idx / Kpad, c = idx % Kpad;
    Wp[idx] = (r < Nout && c < K) ? W[(size_t)r * K + c] : 0.0f;
}

// ---------------------------------------------------------------------------
// Degree / normalization helpers
// ---------------------------------------------------------------------------
__global__ void k_fill1(float* __restrict__ p, int n) {
    int i = blockIdx.x * blockDim.x + threadIdx.x;
    if (i < n) p[i] = 1.0f;                    // self-loop contributes 1
}

__global__ void k_deg(const int* __restrict__ dst, float* __restrict__ deg, int E) {
    int e = blockIdx.x * blockDim.x + threadIdx.x;
    if (e < E) atomicAdd(&deg[dst[e]], 1.0f);
}

__global__ void k_rsqrt(float* __restrict__ p, int n) {
    int i = blockIdx.x * blockDim.x + threadIdx.x;
    if (i < n) { float d = p[i]; p[i] = (d > 0.0f) ? rsqrtf(d) : 0.0f; }
}

__global__ void k_zero(float* __restrict__ p, size_t n) {
    size_t i = (size_t)blockIdx.x * blockDim.x + threadIdx.x;
    if (i < n) p[i] = 0.0f;
}

// ---------------------------------------------------------------------------
// Edge aggregation: out[dst] += h[src] * dinv[src]*dinv[dst]
// Wave-per-edge (coalesced row gather + f32 atomic scatter), grid-stride,
// with prefetch of the next edge's source row.
// ---------------------------------------------------------------------------
__global__ __launch_bounds__(256) void k_agg_wave(
    const float* __restrict__ H, const int* __restrict__ src,
    const int* __restrict__ dst, const float* __restrict__ dinv,
    float* __restrict__ out, int E, int F)
{
    const int lane   = threadIdx.x & 31;
    const int wid    = (int)((blockIdx.x * blockDim.x + threadIdx.x) >> 5);
    const int stride = (int)((gridDim.x * blockDim.x) >> 5);

    for (int e = wid; e < E; e += stride) {
        const int s = src[e];
        const int d = dst[e];
        const int e2 = e + stride;
        if (e2 < E) {  // warm L2/L0 for the next gather
            __builtin_prefetch(&H[(size_t)src[e2] * F], 0, 1);
        }
        const float nrm = dinv[s] * dinv[d];
        const float* __restrict__ hrow = H + (size_t)s * F;
        float* __restrict__ orow = out + (size_t)d * F;
        for (int f = lane; f < F; f += 32) {
            atomicAdd(&orow[f], hrow[f] * nrm);
        }
    }
}

// Thread-per-(edge,feature) variant for very narrow layers (F = 3).
__global__ void k_agg_small(
    const float* __restrict__ H, const int* __restrict__ src,
    const int* __restrict__ dst, const float* __restrict__ dinv,
    float* __restrict__ out, int E, int F)
{
    size_t idx = (size_t)blockIdx.x * blockDim.x + threadIdx.x;
    size_t tot = (size_t)E * F;
    if (idx >= tot) return;
    int e = (int)(idx / F);
    int f = (int)(idx % F);
    int s = src[e], d = dst[e];
    float nrm = dinv[s] * dinv[d];
    atomicAdd(&out[(size_t)d * F + f], H[(size_t)s * F + f] * nrm);
}

// ---------------------------------------------------------------------------
// Epilogue: dest = act( dest + H*dinv^2 (self loop) + bias )
// ---------------------------------------------------------------------------
__global__ void k_finish(float* __restrict__ dest, const float* __restrict__ H,
                         const float* __restrict__ dinv, const float* __restrict__ bias,
                         int N, int F, int doRelu)
{
    size_t idx = (size_t)blockIdx.x * blockDim.x + threadIdx.x;
    size_t tot = (size_t)N * F;
    if (idx >= tot) return;
    int i = (int)(idx / F);
    int f = (int)(idx % F);
    float di = dinv[i];
    float v  = dest[idx] + H[idx] * di * di + bias[f];
    dest[idx] = doRelu ? fmaxf(v, 0.0f) : v;
}

// ---------------------------------------------------------------------------
extern "C" void kernel_launch(void* const* d_in, const int* in_sizes, int n_in,
                              void* d_out, int out_size, void* d_ws, size_t ws_size,
                              hipStream_t stream) {
    (void)n_in; (void)out_size; (void)ws_size;

    const float* x   = (const float*)d_in[0];
    const int*   ei  = (const int*)  d_in[1];
    const float* W1  = (const float*)d_in[2];
    const float* b1  = (const float*)d_in[3];
    const float* W1b = (const float*)d_in[4];
    const float* b1b = (const float*)d_in[5];
    const float* W2b = (const float*)d_in[6];
    const float* b2b = (const float*)d_in[7];
    const float* W2  = (const float*)d_in[8];
    const float* b2  = (const float*)d_in[9];

    const int F_IN = 500, H1 = 200, H2 = 100, H3 = 40, C = 3;
    const int N = in_sizes[0] / F_IN;
    const int E = in_sizes[1] / 2;
    const int* srcIdx = ei;          // edge_index[0]
    const int* dstIdx = ei + E;      // edge_index[1]

    // workspace layout: dinv [N] | Wp [256*512] | bufH [N*H1] | bufA [N*H1]
    char*  ws   = (char*)d_ws;
    float* dinv = (float*)ws;
    size_t off  = ((size_t)N * sizeof(float) + 511) & ~(size_t)511;
    float* Wpad = (float*)(ws + off);
    off += (size_t)256 * 512 * sizeof(float);
    float* bufH = (float*)(ws + off);
    off += (((size_t)N * H1 * sizeof(float)) + 511) & ~(size_t)511;
    float* bufA = (float*)(ws + off);
    float* out  = (float*)d_out;

    // ---- degree + symmetric normalization --------------------------------
    k_fill1<<<(N + 255) / 256, 256, 0, stream>>>(dinv, N);
    k_deg  <<<(E + 255) / 256, 256, 0, stream>>>(dstIdx, dinv, E);
    k_rsqrt<<<(N + 255) / 256, 256, 0, stream>>>(dinv, N);

    const int gemmBlocks = (N + 15) / 16;
    const int aggBlocks  = 65536;    // grid-stride, ~3 edges/wave, enables prefetch

    auto layer = [&](const float* in, const float* Wt, const float* bias,
                     float* dest, int K, int F, int relu) {
        const int Kpad = (K + TK - 1) & ~(TK - 1);
        k_padW<<<(256 * Kpad + 255) / 256, 256, 0, stream>>>(Wt, Wpad, F, K, Kpad);
        k_gemm_wmma<<<gemmBlocks, 256, 0, stream>>>(in, Wpad, bufH, N, K, Kpad, F);
        size_t tot = (size_t)N * F;
        k_zero<<<(int)((tot + 255) / 256), 256, 0, stream>>>(dest, tot);
        if (F >= 32) {
            k_agg_wave<<<aggBlocks, 256, 0, stream>>>(bufH, srcIdx, dstIdx, dinv, dest, E, F);
        } else {
            size_t tote = (size_t)E * F;
            k_agg_small<<<(int)((tote + 255) / 256), 256, 0, stream>>>(
                bufH, srcIdx, dstIdx, dinv, dest, E, F);
        }
        k_finish<<<(int)((tot + 255) / 256), 256, 0, stream>>>(
            dest, bufH, dinv, bias, N, F, relu);
    };

    layer(x,    W1,  b1,  bufA, F_IN, H1, 1);
    layer(bufA, W1b, b1b, bufA, H1,   H2, 1);   // gemm consumes bufA before zero
    layer(bufA, W2b, b2b, bufA, H2,   H3, 1);
    layer(bufA, W2,  b2,  out,  H3,   C,  0);
}